// DisentangledGNN_83030307766551
// MI455X (gfx1250) — compile-verified
//
#include <hip/hip_runtime.h>
#include <hip/hip_bf16.h>

#define DFEAT 128
#define LDSS  132   // padded LDS row stride (floats) to dodge bank conflicts

typedef float v2f __attribute__((ext_vector_type(2)));
typedef float v8f __attribute__((ext_vector_type(8)));

// ---------------------------------------------------------------------------
// deg init: deg = 1.0 for 3*N entries
// ---------------------------------------------------------------------------
__global__ void init_ones_kernel(float* __restrict__ p, int n) {
    int i = blockIdx.x * blockDim.x + threadIdx.x;
    if (i < n) p[i] = 1.0f;
}

// ---------------------------------------------------------------------------
// deg[c*N + tgt[e]] += alpha_c[e]*w_c[e]   (layer-invariant)
// ---------------------------------------------------------------------------
__global__ void deg_scatter_kernel(const int* __restrict__ tgt,
                                   const float* __restrict__ a0, const float* __restrict__ a1,
                                   const float* __restrict__ a2,
                                   const float* __restrict__ w0, const float* __restrict__ w1,
                                   const float* __restrict__ w2,
                                   float* __restrict__ deg, int N, int E) {
    int e = blockIdx.x * blockDim.x + threadIdx.x;
    if (e >= E) return;
    int t = tgt[e];
    atomicAdd(&deg[t],         a0[e] * w0[e]);
    atomicAdd(&deg[N + t],     a1[e] * w1[e]);
    atomicAdd(&deg[2 * N + t], a2[e] * w2[e]);
}

// ---------------------------------------------------------------------------
// dinv[i] = pow(max(deg[i],1e-6), deg_power[l*3 + (i/N)])
// ---------------------------------------------------------------------------
__global__ void dinv_kernel(const float* __restrict__ deg,
                            const float* __restrict__ deg_power, int layer,
                            float* __restrict__ dinv, int N) {
    int i = blockIdx.x * blockDim.x + threadIdx.x;
    if (i >= 3 * N) return;
    int c = i / N;
    float d = fmaxf(deg[i], 1e-6f);
    dinv[i] = powf(d, deg_power[layer * 3 + c]);
}

// ---------------------------------------------------------------------------
// norm[c*E+e] = alpha_c*w_c * dinv_c[tgt] * dinv_c[src]
// ---------------------------------------------------------------------------
__global__ void norm_kernel(const int* __restrict__ src, const int* __restrict__ tgt,
                            const float* __restrict__ a0, const float* __restrict__ a1,
                            const float* __restrict__ a2,
                            const float* __restrict__ w0, const float* __restrict__ w1,
                            const float* __restrict__ w2,
                            const float* __restrict__ dinv,
                            float* __restrict__ nrm, int N, int E) {
    int e = blockIdx.x * blockDim.x + threadIdx.x;
    if (e >= E) return;
    int s = src[e], t = tgt[e];
    nrm[e]         = a0[e] * w0[e] * dinv[t]         * dinv[s];
    nrm[E + e]     = a1[e] * w1[e] * dinv[N + t]     * dinv[N + s];
    nrm[2 * E + e] = a2[e] * w2[e] * dinv[2 * N + t] * dinv[2 * N + s];
}

// ---------------------------------------------------------------------------
// GEMM: OUT = X @ W (+bias for the W_self slot). grid.y in {0,1,2}: W3[c] -> h_c,
// grid.y==3: W_self -> acc (+bias). One block = 16 rows x 128 cols; 8 waves,
// each wave owns one 16x16 tile and runs 32 x v_wmma_f32_16x16x4_f32 along K.
// LDS: X tile (16x128) + W transposed (128x128), padded stride.
// ---------------------------------------------------------------------------
__global__ __launch_bounds__(256) void gemm_wmma_kernel(
    const float* __restrict__ X,      // [N,128]
    const float* __restrict__ W3,     // [3,128,128] (this layer)
    const float* __restrict__ Wself,  // [128,128]   (this layer)
    const float* __restrict__ bias,   // [128]       (this layer)
    float* __restrict__ h0, float* __restrict__ h1, float* __restrict__ h2,
    float* __restrict__ acc, int N) {
    extern __shared__ float smem[];
    float* sX = smem;                   // 16 * LDSS
    float* sW = smem + 16 * LDSS;       // 128 * LDSS, stored transposed: sW[col*LDSS + k]

    const int w = blockIdx.y;
    const float* W = (w < 3) ? (W3 + (size_t)w * DFEAT * DFEAT) : Wself;
    float* OUT = (w == 0) ? h0 : (w == 1) ? h1 : (w == 2) ? h2 : acc;

    const int row0 = blockIdx.x * 16;
    const int tid  = threadIdx.x;

    // stage X tile (16x128) as float4
    for (int i = tid; i < 16 * 32; i += 256) {
        int r  = i >> 5;
        int c4 = i & 31;
        int gr = row0 + r; if (gr >= N) gr = N - 1;
        float4 v = ((const float4*)(X + (size_t)gr * DFEAT))[c4];
        float* dst = sX + r * LDSS + c4 * 4;
        dst[0] = v.x; dst[1] = v.y; dst[2] = v.z; dst[3] = v.w;
    }
    // stage W transposed (128x128)
    for (int i = tid; i < 128 * 32; i += 256) {
        int k  = i >> 5;
        int c4 = i & 31;
        float4 v = ((const float4*)(W + (size_t)k * DFEAT))[c4];
        int c = c4 * 4;
        sW[(c + 0) * LDSS + k] = v.x;
        sW[(c + 1) * LDSS + k] = v.y;
        sW[(c + 2) * LDSS + k] = v.z;
        sW[(c + 3) * LDSS + k] = v.w;
    }
    __syncthreads();

    const int wave = tid >> 5;      // 0..7 -> 16-col tile
    const int lane = tid & 31;
    const int half = lane >> 4;     // 0/1
    const int m    = lane & 15;
    const int col0 = wave * 16;

    const float* xrow = sX + m * LDSS;             // A row m of tile
    const float* wcol = sW + (col0 + m) * LDSS;    // B col (col0+m)

    v8f c = {};
#pragma unroll
    for (int k0 = 0; k0 < DFEAT; k0 += 4) {
        // A 16x4 f32 layout: lanes 0-15 hold K=0,1 ; lanes 16-31 hold K=2,3
        v2f a, b;
        a[0] = xrow[k0 + 2 * half + 0];
        a[1] = xrow[k0 + 2 * half + 1];
        // B 4x16 f32: VGPR v holds row K = v + 2*half, striped across lanes (col=m)
        b[0] = wcol[k0 + 2 * half + 0];
        b[1] = wcol[k0 + 2 * half + 1];
        c = __builtin_amdgcn_wmma_f32_16x16x4_f32(
                /*neg_a=*/false, a, /*neg_b=*/false, b,
                /*c_mod=*/(short)0, c, /*reuse_a=*/false, /*reuse_b=*/false);
    }

    // C/D 16x16 f32 layout: VGPR v -> row (v + 8*half), col = m
    const float badd = (w == 3) ? bias[col0 + m] : 0.0f;
#pragma unroll
    for (int v = 0; v < 8; ++v) {
        int gr = row0 + v + 8 * half;
        if (gr < N) OUT[(size_t)gr * DFEAT + col0 + m] = c[v] + badd;
    }
}

// ---------------------------------------------------------------------------
// Fused 3-channel edge message scatter:
//   acc[tgt] += norm0*h0[src] + norm1*h1[src] + norm2*h2[src]
// One wave per edge; each lane owns 4 consecutive features (float4 gather,
// 4x global_atomic_add_f32 scatter).
// ---------------------------------------------------------------------------
__global__ __launch_bounds__(256) void edge_scatter_kernel(
    const int* __restrict__ src, const int* __restrict__ tgt,
    const float* __restrict__ nrm,   // [3,E]
    const float* __restrict__ h0, const float* __restrict__ h1,
    const float* __restrict__ h2,
    float* __restrict__ acc, int E) {
    int e = blockIdx.x * 8 + (threadIdx.x >> 5);
    if (e >= E) return;
    int lane = threadIdx.x & 31;
    int s = src[e], t = tgt[e];
    float n0 = nrm[e];
    float n1 = nrm[E + e];
    float n2 = nrm[2 * (size_t)E + e];
    float4 v0 = ((const float4*)(h0 + (size_t)s * DFEAT))[lane];
    float4 v1 = ((const float4*)(h1 + (size_t)s * DFEAT))[lane];
    float4 v2 = ((const float4*)(h2 + (size_t)s * DFEAT))[lane];
    float mx = n0 * v0.x + n1 * v1.x + n2 * v2.x;
    float my = n0 * v0.y + n1 * v1.y + n2 * v2.y;
    float mz = n0 * v0.z + n1 * v1.z + n2 * v2.z;
    float mw = n0 * v0.w + n1 * v1.w + n2 * v2.w;
    float* out = acc + (size_t)t * DFEAT + lane * 4;
    atomicAdd(out + 0, mx);
    atomicAdd(out + 1, my);
    atomicAdd(out + 2, mz);
    atomicAdd(out + 3, mw);
}

// ---------------------------------------------------------------------------
// LayerNorm (over 128 features) + ReLU. One block (128 threads) per row.
// ---------------------------------------------------------------------------
__global__ __launch_bounds__(128) void ln_relu_kernel(
    const float* __restrict__ in, const float* __restrict__ g,
    const float* __restrict__ b, float* __restrict__ out, int N) {
    int row = blockIdx.x;
    if (row >= N) return;
    int f = threadIdx.x;
    float v = in[(size_t)row * DFEAT + f];
    __shared__ float red[DFEAT];
    red[f] = v;
    __syncthreads();
    for (int s = 64; s > 0; s >>= 1) {
        if (f < s) red[f] += red[f + s];
        __syncthreads();
    }
    float mu = red[0] * (1.0f / DFEAT);
    __syncthreads();
    float d = v - mu;
    red[f] = d * d;
    __syncthreads();
    for (int s = 64; s > 0; s >>= 1) {
        if (f < s) red[f] += red[f + s];
        __syncthreads();
    }
    float var = red[0] * (1.0f / DFEAT);
    float y = d * rsqrtf(var + 1e-5f) * g[f] + b[f];
    out[(size_t)row * DFEAT + f] = fmaxf(y, 0.0f);
}

// ---------------------------------------------------------------------------
extern "C" void kernel_launch(void* const* d_in, const int* in_sizes, int n_in,
                              void* d_out, int out_size, void* d_ws, size_t ws_size,
                              hipStream_t stream) {
    const float* x    = (const float*)d_in[0];
    const int*   ei   = (const int*)d_in[1];
    const float* a0   = (const float*)d_in[2];
    const float* a1   = (const float*)d_in[3];
    const float* a2   = (const float*)d_in[4];
    const float* w0   = (const float*)d_in[5];
    const float* w1   = (const float*)d_in[6];
    const float* w2   = (const float*)d_in[7];
    const float* W3   = (const float*)d_in[8];    // [L,3,128,128]
    const float* Ws   = (const float*)d_in[9];    // [L,128,128]
    const float* bias = (const float*)d_in[10];   // [L,128]
    const float* dpow = (const float*)d_in[11];   // [L,3]
    const float* lng  = (const float*)d_in[12];   // [128]
    const float* lnb  = (const float*)d_in[13];   // [128]

    const int N = in_sizes[0] / DFEAT;
    const int E = in_sizes[2];
    const int L = in_sizes[9] / (DFEAT * DFEAT);
    const int* srcp = ei;       // edge_index[0]
    const int* tgtp = ei + E;   // edge_index[1]

    float* ws = (float*)d_ws;
    size_t off = 0;
    float* deg  = ws + off; off += (size_t)3 * N;
    float* dinv = ws + off; off += (size_t)3 * N;
    float* nrm  = ws + off; off += (size_t)3 * E;
    float* h0   = ws + off; off += (size_t)N * DFEAT;
    float* h1   = ws + off; off += (size_t)N * DFEAT;
    float* h2   = ws + off; off += (size_t)N * DFEAT;
    float* xbuf = ws + off; off += (size_t)N * DFEAT;
    float* accb = ws + off; off += (size_t)N * DFEAT;

    // degree (layer-invariant)
    init_ones_kernel<<<(3 * N + 255) / 256, 256, 0, stream>>>(deg, 3 * N);
    deg_scatter_kernel<<<(E + 255) / 256, 256, 0, stream>>>(
        tgtp, a0, a1, a2, w0, w1, w2, deg, N, E);

    const size_t lds_bytes = (size_t)(16 + DFEAT) * LDSS * sizeof(float);
    const float* cur = x;
    for (int l = 0; l < L; ++l) {
        float* acc = (l == L - 1) ? (float*)d_out : accb;
        dinv_kernel<<<(3 * N + 255) / 256, 256, 0, stream>>>(deg, dpow, l, dinv, N);
        norm_kernel<<<(E + 255) / 256, 256, 0, stream>>>(
            srcp, tgtp, a0, a1, a2, w0, w1, w2, dinv, nrm, N, E);
        dim3 grid((N + 15) / 16, 4);
        gemm_wmma_kernel<<<grid, 256, lds_bytes, stream>>>(
            cur, W3 + (size_t)l * 3 * DFEAT * DFEAT, Ws + (size_t)l * DFEAT * DFEAT,
            bias + (size_t)l * DFEAT, h0, h1, h2, acc, N);
        edge_scatter_kernel<<<(E + 7) / 8, 256, 0, stream>>>(
            srcp, tgtp, nrm, h0, h1, h2, acc, E);
        if (l < L - 1) {
            ln_relu_kernel<<<N, DFEAT, 0, stream>>>(acc, lng, lnb, xbuf, N);
            cur = xbuf;
        }
    }
}